// GConvSE3_59407987638994
// MI455X (gfx1250) — compile-verified
//
#include <hip/hip_runtime.h>

#define E_EDGES 65536
#define N_NODES 4096
#define MIDW    32
#define EPSB    1e-5f

typedef __attribute__((ext_vector_type(2))) float v2f;
typedef __attribute__((ext_vector_type(8))) float v8f;

// ---------------- workspace layout (float offsets) ----------------
#define WS_STATSR 0                         // 2 floats: sum(r), sum(r^2)
#define WS_CNT    16                        // N floats: in-degree counts
#define WS_SUMX   (WS_CNT + N_NODES)        // 4x32 sums of x2
#define WS_SUMXX  (WS_SUMX + 128)           // 4x32 sums of x2^2
#define WS_A1     (WS_SUMXX + 128)          // folded layer-1 coeffs
#define WS_B1     (WS_A1 + 128)
#define WS_G2     (WS_B1 + 128)             // folded layer-2 BN affine
#define WS_C2     (WS_G2 + 128)
#define WS_S0     8192                      // node self terms deg0: N*16
#define WS_S1     (WS_S0 + N_NODES*16)      // node self terms deg1: N*48
#define WS_M0     (WS_S1 + N_NODES*48)      // msg accumulator deg0: N*16
#define WS_M1     (WS_M0 + N_NODES*16)      // msg accumulator deg1: N*48
#define WS_W3X    (WS_M1 + N_NODES*48)      // fragment-transposed w3: 49152
#define WS_B3X    (WS_W3X + 49152)          // f-major b3_11: 768
#define WS_W2X    (WS_B3X + 768)            // fragment-transposed w2: 4x1024

__device__ __forceinline__ v8f wmma_f32(v2f a, v2f b, v8f c) {
  // V_WMMA_F32_16X16X4_F32 : D = A(16x4 f32) x B(4x16 f32) + C(16x16 f32)
  return __builtin_amdgcn_wmma_f32_16x16x4_f32(false, a, false, b, (short)0, c,
                                               false, false);
}

// Swap data between the two 16-lane rows of a wave32 (xor-16) as pure VALU.
__device__ __forceinline__ float half_swap(float v) {
#if __has_builtin(__builtin_amdgcn_permlanex16)
  int i = __builtin_bit_cast(int, v);
  i = __builtin_amdgcn_permlanex16(i, i, 0x76543210, 0xfedcba98, true, false);
  return __builtin_bit_cast(float, i);
#else
  return __shfl_xor(v, 16, 32);
#endif
}

// ---------------- utility kernels ----------------

__global__ void k_zero(float* ws) {
  const int stride = gridDim.x * blockDim.x;
  int t = blockIdx.x * blockDim.x + threadIdx.x;
  for (int i = t; i < 8192; i += stride) ws[i] = 0.f;                 // stats/cnt/coeffs
  for (int i = t; i < N_NODES * 64; i += stride) ws[WS_M0 + i] = 0.f; // msg accumulators
}

__global__ void k_rstats(const float* __restrict__ r,
                         const int* __restrict__ edst, float* ws) {
  const int e = blockIdx.x * blockDim.x + threadIdx.x;
  const float rv = r[e];
  atomicAdd(&ws[WS_CNT + edst[e]], 1.f);
  float s1 = rv, s2 = rv * rv;
  for (int m = 16; m > 0; m >>= 1) {
    s1 += __shfl_xor(s1, m, 32);
    s2 += __shfl_xor(s2, m, 32);
  }
  if ((threadIdx.x & 31) == 0) {
    atomicAdd(&ws[WS_STATSR + 0], s1);
    atomicAdd(&ws[WS_STATSR + 1], s2);
  }
}

__global__ void k_a1b1(const float* __restrict__ rw1, const float* __restrict__ rg1,
                       const float* __restrict__ rbe1, float* ws) {
  const int t = threadIdx.x;  // 128 = 4 pairs x 32
  const float mr = ws[WS_STATSR + 0] / (float)E_EDGES;
  const float vr = ws[WS_STATSR + 1] / (float)E_EDGES - mr * mr;
  const float w = rw1[t];
  const float a = w * rg1[t] * rsqrtf(vr * w * w + EPSB);
  ws[WS_A1 + t] = a;
  ws[WS_B1 + t] = rbe1[t] - mr * a;
}

__global__ void k_g2c2(const float* __restrict__ rg2, const float* __restrict__ rbe2,
                       float* ws) {
  const int t = threadIdx.x;  // 128
  const float mu = ws[WS_SUMX + t] / (float)E_EDGES;
  const float var = ws[WS_SUMXX + t] / (float)E_EDGES - mu * mu;
  const float g = rg2[t] * rsqrtf(var + EPSB);
  ws[WS_G2 + t] = g;
  ws[WS_C2 + t] = rbe2[t] - mu * g;
}

__global__ void k_self(const float* __restrict__ h0, const float* __restrict__ h1,
                       const float* __restrict__ W0, const float* __restrict__ W1,
                       float* ws) {
  const int t = blockIdx.x * blockDim.x + threadIdx.x;  // N*16
  const int n = t >> 4, m = t & 15;
  float s0 = 0.f;
#pragma unroll
  for (int i = 0; i < 16; i++) s0 += W0[m * 16 + i] * h0[n * 16 + i];
  ws[WS_S0 + t] = s0;
#pragma unroll
  for (int p = 0; p < 3; p++) {
    float s = 0.f;
#pragma unroll
    for (int i = 0; i < 16; i++) s += W1[m * 16 + i] * h1[(n * 16 + i) * 3 + p];
    ws[WS_S1 + t * 3 + p] = s;
  }
}

// Re-layout w3/w2 so B/A-fragments are contiguous 128-bit loads:
//   wx[col*32 + ((k>>1)&1)*16 + (k>>2)*2 + (k&1)] = w[k*NC + col]
// Per lane: half h reads wx[col*32 + h*16 .. +15] -> 4x global_load_b128.
// Also b3_11 into f-major order so bias seeds are contiguous.
__global__ void k_w3t(const float* __restrict__ w0, const float* __restrict__ w1,
                      const float* __restrict__ w2, const float* __restrict__ w3,
                      const float* __restrict__ b11, const float* __restrict__ rw2,
                      float* ws) {
  const int t = blockIdx.x * blockDim.x + threadIdx.x;
  if (t < 49152) {
    int local, NC, base;
    const float* src;
    if (t < 8192)       { local = t;         NC = 256; src = w0; base = 0; }
    else if (t < 16384) { local = t - 8192;  NC = 256; src = w1; base = 8192; }
    else if (t < 24576) { local = t - 16384; NC = 256; src = w2; base = 16384; }
    else                { local = t - 24576; NC = 768; src = w3; base = 24576; }
    const int col = local >> 5, k = local & 31;
    const int dk = ((k >> 1) & 1) * 16 + (k >> 2) * 2 + (k & 1);
    ws[WS_W3X + base + col * 32 + dk] = src[k * NC + col];
  } else if (t < 49152 + 768) {
    const int i = t - 49152;  // b3_11 flat index m*48 + ii*3 + f
    const int m = i / 48, rr = i % 48, ii = rr / 3, f = rr % 3;
    ws[WS_B3X + f * 256 + m * 16 + ii] = b11[i];
  } else if (t < 49152 + 768 + 4096) {
    const int i = t - 49152 - 768;  // p*1024 + k*32 + j
    const int p = i >> 10, k = (i >> 5) & 31, j = i & 31;
    const int dj = ((j >> 1) & 1) * 16 + (j >> 2) * 2 + (j & 1);
    ws[WS_W2X + p * 1024 + k * 32 + dj] = rw2[(p * 32 + j) * 32 + k];
  }
}

// Load 8 WMMA operand fragments (one 16x32 operand half) from a dk-permuted
// weight row: 4x b128.
__device__ __forceinline__ void load_wfrags(const float* __restrict__ wrow,
                                            v2f (&F)[8]) {
  const float4* wp = (const float4*)wrow;
  const float4 a = wp[0], b = wp[1], c = wp[2], d = wp[3];
  F[0].x = a.x; F[0].y = a.y; F[1].x = a.z; F[1].y = a.w;
  F[2].x = b.x; F[2].y = b.y; F[3].x = b.z; F[3].y = b.w;
  F[4].x = c.x; F[4].y = c.y; F[5].x = c.z; F[5].y = c.w;
  F[6].x = d.x; F[6].y = d.y; F[7].x = d.z; F[7].y = d.w;
}

// Load 8 fragments of an LDS [32]-float activation row (k = 4s+2h, +1).
__device__ __forceinline__ void load_afrags(const float* __restrict__ row, int half,
                                            v2f (&F)[8]) {
  const float2* yp = (const float2*)row;
#pragma unroll
  for (int s = 0; s < 8; s++) {
    const float2 y = yp[2 * s + half];
    F[s].x = y.x;
    F[s].y = y.y;
  }
}

// ---------------- x2 batch-stats kernel (WMMA) ----------------
// x2 = y1 @ w2 + b2 per pair; accumulate per-channel sum and sum-of-squares.
__global__ __launch_bounds__(64) void k_x2stats(const float* __restrict__ r,
                                                const float* __restrict__ rb2,
                                                float* ws) {
  __shared__ __align__(16) float y1s[4][32][32];
  const int tid = threadIdx.x;
  const int lane = tid & 31;
  const int n16 = lane & 15, half = lane >> 4;
  const int waveBase = (tid >> 5) * 16;

  float sx[4][2], sxx[4][2];
#pragma unroll
  for (int p = 0; p < 4; p++) { sx[p][0] = sx[p][1] = sxx[p][0] = sxx[p][1] = 0.f; }

  for (int chunk = 0; chunk < 16; chunk++) {
    const int base = blockIdx.x * 512 + chunk * 32;
    __syncthreads();  // protect y1s reuse across chunks
    {
      const int t = tid & 31;
      const float rv = r[base + t];
      const int p0 = (tid < 32) ? 0 : 2;
      for (int pp = 0; pp < 2; pp++) {
        const int p = p0 + pp;
#pragma unroll
        for (int j = 0; j < MIDW; j++)
          y1s[p][t][j] = fmaxf(ws[WS_A1 + p * 32 + j] * rv + ws[WS_B1 + p * 32 + j], 0.f);
      }
    }
    __syncthreads();
#pragma unroll
    for (int p = 0; p < 4; p++) {
      v2f aF[8];
      load_afrags(&y1s[p][waveBase + n16][0], half, aF);
#pragma unroll
      for (int kt = 0; kt < 2; kt++) {
        const int kcol = kt * 16 + n16;
        const float bb = rb2[p * 32 + kcol];
        v8f acc = {bb, bb, bb, bb, bb, bb, bb, bb};
        v2f bF[8];
        load_wfrags(ws + WS_W2X + p * 1024 + kcol * 32 + half * 16, bF);
#pragma unroll
        for (int s = 0; s < 8; s++) acc = wmma_f32(aF[s], bF[s], acc);
        float s1 = acc[0], s2 = acc[0] * acc[0];
#pragma unroll
        for (int j = 1; j < 8; j++) {
          s1 += acc[j];
          s2 = fmaf(acc[j], acc[j], s2);
        }
        s1 += half_swap(s1);
        s2 += half_swap(s2);
        sx[p][kt] += s1;
        sxx[p][kt] += s2;
      }
    }
  }
#pragma unroll
  for (int p = 0; p < 4; p++)
#pragma unroll
    for (int kt = 0; kt < 2; kt++) {
      const int kcol = kt * 16 + n16;
      if (half == 0) atomicAdd(&ws[WS_SUMX + p * 32 + kcol], sx[p][kt]);
      else           atomicAdd(&ws[WS_SUMXX + p * 32 + kcol], sxx[p][kt]);
    }
}

// ---------------- main fused edge kernel ----------------

struct GParams {
  const float *h0, *h1, *r, *b00, *b01, *b10, *b11, *rb2;
  const float *b3_0, *b3_1, *b3_2;
  const int *esrc, *edst;
  float* ws;
};

// Phase-1b: y2 = relu((y1 @ w2 + b2) * G2 + C2) for this wave's 16 edges, via WMMA.
__device__ __forceinline__ void compute_y2_wmma(
    const float* __restrict__ y1row,  // &y1s[p][waveBase + n16][0]
    const float* __restrict__ w2xp,   // ws + WS_W2X + p*1024
    const float* __restrict__ rb2p,   // rb2 + p*32
    const float* __restrict__ ws, int pairOff,
    float* __restrict__ y2dst,        // &y2s[p][waveBase][0]
    int n16, int half) {
  v2f aF[8];
  load_afrags(y1row, half, aF);
#pragma unroll
  for (int kt = 0; kt < 2; kt++) {
    const int kcol = kt * 16 + n16;
    const float bb = rb2p[kcol];
    v8f acc = {bb, bb, bb, bb, bb, bb, bb, bb};
    v2f bF[8];
    load_wfrags(w2xp + kcol * 32 + half * 16, bF);
#pragma unroll
    for (int s = 0; s < 8; s++) acc = wmma_f32(aF[s], bF[s], acc);
    const float g = ws[WS_G2 + pairOff + kcol];
    const float cc = ws[WS_C2 + pairOff + kcol];
#pragma unroll
    for (int j = 0; j < 8; j++)
      y2dst[(j + 8 * half) * 32 + kcol] = fmaxf(fmaf(acc[j], g, cc), 0.f);
  }
}

// Transposed-GEMM pair processor.  One wave owns 16 edges.
//   D[row, edge] = sum_k w3[k, wcol(row)] * y2[edge, k]  (+ b3 bias via C seed)
// D layout: vgpr j, half h -> (i = j + 8h), lane%16 -> edge.  Contraction with
// the premixed u-tile is 8 per-lane FMAs + one half_swap.
template <int PAIR>
__device__ __forceinline__ void process_pair(
    const float* __restrict__ y2row,   // &y2s[PAIR][waveBase + n16][0]
    const float* __restrict__ uBase,   // &us[0][waveBase + n16][0]
    const float* __restrict__ w3x,     // transposed w3 region for this pair
    const float* __restrict__ b3e,     // bias (pair3: f-major b3x)
    int n16, int half,
    float (&msg0T)[16], float (&msg1T)[3][16]) {
  constexpr int NF = (PAIR == 3) ? 3 : 1;
  constexpr int NV = (PAIR == 1 || PAIR == 3) ? 3 : 1;
  constexpr bool TOM1 = (PAIR == 1 || PAIR == 3);

  // B fragments: y2^T, hoisted across all column tiles (LDS b64 loads).
  v2f bF[8];
  load_afrags(y2row, half, bF);

  for (int f = 0; f < NF; f++) {
    const int vbase = (PAIR == 0) ? 0 : (PAIR == 2) ? 1 : (PAIR == 1) ? 2 : (5 + 3 * f);
    float uT[NV][8];
#pragma unroll
    for (int v = 0; v < NV; v++) {
      const float4* up = (const float4*)(uBase + (vbase + v) * 512 + 8 * half);
      const float4 uA = up[0], uB = up[1];
      uT[v][0] = uA.x; uT[v][1] = uA.y; uT[v][2] = uA.z; uT[v][3] = uA.w;
      uT[v][4] = uB.x; uT[v][5] = uB.y; uT[v][6] = uB.z; uT[v][7] = uB.w;
    }
    const float* bias = b3e + ((PAIR == 3) ? f * 256 : 0);

#pragma unroll
    for (int c = 0; c < 16; c++) {
      // bias seed (contiguous, 2x b128)
      const float4* bp = (const float4*)(bias + c * 16 + 8 * half);
      const float4 bA = bp[0], bB = bp[1];
      v8f acc;
      acc[0] = bA.x; acc[1] = bA.y; acc[2] = bA.z; acc[3] = bA.w;
      acc[4] = bB.x; acc[5] = bB.y; acc[6] = bB.z; acc[7] = bB.w;

      // A fragments for this column tile: 4x b128 from w3x
      const int wcol = (PAIR == 3) ? (c * 48 + n16 * 3 + f) : (c * 16 + n16);
      v2f aF[8];
      load_wfrags(w3x + wcol * 32 + half * 16, aF);

#pragma unroll
      for (int s = 0; s < 8; s++) acc = wmma_f32(aF[s], bF[s], acc);

#pragma unroll
      for (int v = 0; v < NV; v++) {
        float part = acc[0] * uT[v][0];
#pragma unroll
        for (int j = 1; j < 8; j++) part = fmaf(acc[j], uT[v][j], part);
        part += half_swap(part);  // both halves now hold the full sum over i
        if (TOM1) msg1T[v][c] += part; else msg0T[c] += part;
      }
    }
  }
}

__global__ __launch_bounds__(64) void k_main(GParams P) {
  __shared__ __align__(16) float y1s[4][32][32];  // layer-1 activations
  __shared__ __align__(16) float y2s[4][32][32];  // radial activations per pair
  __shared__ __align__(16) float us[14][32][16];  // basis-premixed gathered feats
  const int tid = threadIdx.x;
  const int tileBase = blockIdx.x * 32;
  float* ws = P.ws;

  const int lane = tid & 31;
  const int waveBase = (tid >> 5) * 16;
  const int n16 = lane & 15;
  const int half = lane >> 4;

  // ---- phase 1a: y1 (wave0: pairs 0,1 + u0 premix; wave1: pairs 2,3 + u1 premix)
  {
    const int t = tid & 31;
    const int e = tileBase + t;
    const float rv = P.r[e];
    const int p0 = (tid < 32) ? 0 : 2;
    for (int pp = 0; pp < 2; pp++) {
      const int p = p0 + pp;
#pragma unroll
      for (int j = 0; j < MIDW; j++)
        y1s[p][t][j] = fmaxf(ws[WS_A1 + p * 32 + j] * rv + ws[WS_B1 + p * 32 + j], 0.f);
    }
    const int sN = P.esrc[e];
    if (tid < 32) {
      float u0[16];
#pragma unroll
      for (int i = 0; i < 16; i++) u0[i] = P.h0[sN * 16 + i];
      const float c00 = P.b00[e];
#pragma unroll
      for (int i = 0; i < 16; i++) us[0][t][i] = c00 * u0[i];
#pragma unroll
      for (int p = 0; p < 3; p++) {
        const float cp = P.b01[e * 3 + p];
#pragma unroll
        for (int i = 0; i < 16; i++) us[2 + p][t][i] = cp * u0[i];
      }
    } else {
      float u1a[16], u1b[16], u1c[16];
#pragma unroll
      for (int i = 0; i < 16; i++) {
        u1a[i] = P.h1[(sN * 16 + i) * 3 + 0];
        u1b[i] = P.h1[(sN * 16 + i) * 3 + 1];
        u1c[i] = P.h1[(sN * 16 + i) * 3 + 2];
      }
      const float q0 = P.b10[e * 3 + 0], q1 = P.b10[e * 3 + 1], q2 = P.b10[e * 3 + 2];
#pragma unroll
      for (int i = 0; i < 16; i++)
        us[1][t][i] = q0 * u1a[i] + q1 * u1b[i] + q2 * u1c[i];
      for (int f = 0; f < 3; f++) {
#pragma unroll
        for (int p = 0; p < 3; p++) {
          const float d0 = P.b11[e * 27 + p * 9 + 0 * 3 + f];
          const float d1 = P.b11[e * 27 + p * 9 + 1 * 3 + f];
          const float d2 = P.b11[e * 27 + p * 9 + 2 * 3 + f];
#pragma unroll
          for (int i = 0; i < 16; i++)
            us[5 + f * 3 + p][t][i] = d0 * u1a[i] + d1 * u1b[i] + d2 * u1c[i];
        }
      }
    }
  }
  __syncthreads();

  // ---- phase 1b: y2 via WMMA (each wave: its own 16 edges, all 4 pairs) ----
  for (int p = 0; p < 4; p++)
    compute_y2_wmma(&y1s[p][waveBase + n16][0], ws + WS_W2X + p * 1024,
                    P.rb2 + p * 32, ws, p * 32, &y2s[p][waveBase][0], n16, half);
  // same-wave LDS writes/reads are in order: no barrier needed before phase 2.

  // ---- phase 2: transposed WMMA GEMMs + per-lane contraction ----
  float msg0T[16];
  float msg1T[3][16];
#pragma unroll
  for (int c = 0; c < 16; c++) {
    msg0T[c] = 0.f; msg1T[0][c] = 0.f; msg1T[1][c] = 0.f; msg1T[2][c] = 0.f;
  }

  const float* uBase = &us[0][waveBase + n16][0];
  process_pair<0>(&y2s[0][waveBase + n16][0], uBase, ws + WS_W3X + 0,     P.b3_0,      n16, half, msg0T, msg1T);
  process_pair<1>(&y2s[1][waveBase + n16][0], uBase, ws + WS_W3X + 8192,  P.b3_1,      n16, half, msg0T, msg1T);
  process_pair<2>(&y2s[2][waveBase + n16][0], uBase, ws + WS_W3X + 16384, P.b3_2,      n16, half, msg0T, msg1T);
  process_pair<3>(&y2s[3][waveBase + n16][0], uBase, ws + WS_W3X + 24576, ws + WS_B3X, n16, half, msg0T, msg1T);

  // ---- phase 3: atomic scatter (halves hold identical copies; split work) ----
  const int e = tileBase + waveBase + n16;
  const int d = P.edst[e];
  if (half == 0) {
#pragma unroll
    for (int c = 0; c < 16; c++) atomicAdd(&ws[WS_M0 + d * 16 + c], msg0T[c]);
#pragma unroll
    for (int c = 0; c < 16; c++) atomicAdd(&ws[WS_M1 + (d * 16 + c) * 3 + 0], msg1T[0][c]);
  } else {
#pragma unroll
    for (int c = 0; c < 16; c++) atomicAdd(&ws[WS_M1 + (d * 16 + c) * 3 + 1], msg1T[1][c]);
#pragma unroll
    for (int c = 0; c < 16; c++) atomicAdd(&ws[WS_M1 + (d * 16 + c) * 3 + 2], msg1T[2][c]);
  }
}

__global__ void k_final(const float* __restrict__ ws, float* __restrict__ out) {
  const int t = blockIdx.x * blockDim.x + threadIdx.x;  // N*64 threads
  if (t < N_NODES * 16) {
    const int n = t >> 4;
    const float c = ws[WS_CNT + n];
    const float cm = fmaxf(c, 1.f);
    out[t] = ws[WS_M0 + t] / cm + ((c > 0.f) ? ws[WS_S0 + t] : 0.f);
  } else {
    const int i = t - N_NODES * 16;
    const int n = i / 48;
    const float c = ws[WS_CNT + n];
    const float cm = fmaxf(c, 1.f);
    out[t] = ws[WS_M1 + i] / cm + ((c > 0.f) ? ws[WS_S1 + i] : 0.f);
  }
}

// ---------------- host entry ----------------

extern "C" void kernel_launch(void* const* d_in, const int* in_sizes, int n_in,
                              void* d_out, int out_size, void* d_ws, size_t ws_size,
                              hipStream_t stream) {
  (void)in_sizes; (void)n_in; (void)out_size; (void)ws_size;
  const float* h0   = (const float*)d_in[0];
  const float* h1   = (const float*)d_in[1];
  const float* r    = (const float*)d_in[2];
  const float* b00  = (const float*)d_in[3];
  const float* b01  = (const float*)d_in[4];
  const float* b10  = (const float*)d_in[5];
  const float* b11  = (const float*)d_in[6];
  const float* rw1  = (const float*)d_in[7];
  // d_in[8] = rb1 (cancels inside BatchNorm; unused)
  const float* rg1  = (const float*)d_in[9];
  const float* rbe1 = (const float*)d_in[10];
  const float* rw2  = (const float*)d_in[11];
  const float* rb2  = (const float*)d_in[12];
  const float* rg2  = (const float*)d_in[13];
  const float* rbe2 = (const float*)d_in[14];
  const float* w3_00 = (const float*)d_in[15];
  const float* b3_00 = (const float*)d_in[16];
  const float* w3_01 = (const float*)d_in[17];
  const float* b3_01 = (const float*)d_in[18];
  const float* w3_10 = (const float*)d_in[19];
  const float* b3_10 = (const float*)d_in[20];
  const float* w3_11 = (const float*)d_in[21];
  const float* b3_11 = (const float*)d_in[22];
  const float* W0   = (const float*)d_in[23];
  const float* W1   = (const float*)d_in[24];
  const int* esrc   = (const int*)d_in[25];
  const int* edst   = (const int*)d_in[26];
  float* ws  = (float*)d_ws;
  float* out = (float*)d_out;

  k_zero<<<256, 256, 0, stream>>>(ws);
  k_w3t<<<(49152 + 768 + 4096 + 255) / 256, 256, 0, stream>>>(
      w3_00, w3_01, w3_10, w3_11, b3_11, rw2, ws);
  k_rstats<<<E_EDGES / 256, 256, 0, stream>>>(r, edst, ws);
  k_a1b1<<<1, 128, 0, stream>>>(rw1, rg1, rbe1, ws);
  k_x2stats<<<E_EDGES / 512, 64, 0, stream>>>(r, rb2, ws);
  k_g2c2<<<1, 128, 0, stream>>>(rg2, rbe2, ws);
  k_self<<<(N_NODES * 16) / 256, 256, 0, stream>>>(h0, h1, W0, W1, ws);

  GParams P;
  P.h0 = h0; P.h1 = h1; P.r = r;
  P.b00 = b00; P.b01 = b01; P.b10 = b10; P.b11 = b11;
  P.rb2 = rb2;
  P.b3_0 = b3_00; P.b3_1 = b3_01; P.b3_2 = b3_10;
  P.esrc = esrc; P.edst = edst; P.ws = ws;
  k_main<<<E_EDGES / 32, 64, 0, stream>>>(P);

  k_final<<<(N_NODES * 64) / 256, 256, 0, stream>>>(ws, out);
}